// SinkhornAutoregressiveFlow_85186381349469
// MI455X (gfx1250) — compile-verified
//
#include <hip/hip_runtime.h>
#include <hip/hip_bf16.h>

// Sinkhorn flow, MI455X (gfx1250) plan:
//  - logits[b,l] = W[tokens[b,l]] exactly (one-hot matmul == row gather), and the
//    output is column t of sinkhorn(W[t].reshape(256,256)). So run ONE block per
//    token value t (256 blocks), not per (b,l): 4x less sinkhorn work than naive
//    and bit-identical math.
//  - 256x256 f32 matrix lives in LDS (~265KB < 320KB/WGP, a CDNA5-only capacity).
//    Row stride 264 floats: 264*4 % 16 == 0 so every row chunk is 16B-aligned
//    (required for b128 async LDS writes); 264 % 64 == 8 makes the column pass
//    bank-conflict-free and the row pass ~2-way (same as the 257 layout).
//  - Stage-in uses CDNA5 GLOBAL_LOAD_ASYNC_TO_LDS_B128 (memory->LDS direct, no
//    VGPR round trip, ASYNCcnt-tracked), finished with s_wait_asynccnt.
//  - Row/col sum-of-exp reductions run on the matrix pipe via V_WMMA_F32_16X16X4_F32
//    against an all-ones B (D[m][n] = rowsum[m] for every n), co-executing with the
//    v_exp_f32 TRANS work. f32 WMMA keeps full precision (no f16 rounding drift).
//  - HBM traffic is only W (64MB, L2-resident at 192MB) + 1MB out: compute-bound,
//    so the effort goes into LDS residency + TRANS/WMMA/LDS pipe overlap.

#define V_DIM   256
#define STRIDE  264          // 16B-aligned rows for b128 async; cols conflict-free
#define N_ITERS 20
#define BL      1024         // B * L = 8 * 128

typedef __attribute__((ext_vector_type(2))) float v2f;
typedef __attribute__((ext_vector_type(8))) float v8f;

__global__ __launch_bounds__(256)
void sinkhorn_per_token_kernel(const int* __restrict__ tokens,
                               const float* __restrict__ W,
                               float* __restrict__ out)
{
    extern __shared__ float smem[];
    float* la  = smem;                       // V_DIM * STRIDE floats (padded matrix)
    float* red = smem + V_DIM * STRIDE;      // V_DIM floats (per-row/col sum staging)

    const int t   = blockIdx.x;              // token value this block solves for
    const int tid = threadIdx.x;

    // ---- Stage W[t] (256x256 row-major) into padded LDS via async b128 copies.
    // No static __shared__ exists, so the dynamic LDS segment starts at offset 0
    // and LDS byte offsets can be computed arithmetically.
    {
        const uint64_t wbase = (uint64_t)(uintptr_t)(W + (size_t)t * (V_DIM * V_DIM));
        for (int j = 0; j < V_DIM * V_DIM; j += 256 * 4) {
            int idx = j + tid * 4;                   // 4 consecutive floats, one row
            int r = idx >> 8, c = idx & 255;
            unsigned ldsOff = (unsigned)((r * STRIDE + c) * 4);  // 16B-aligned
            unsigned gOff   = (unsigned)(idx * 4);               // 16B-aligned
            asm volatile("global_load_async_to_lds_b128 %0, %1, %2"
                         :: "v"(ldsOff), "v"(gOff), "s"(wbase)
                         : "memory");
        }
        asm volatile("s_wait_asynccnt 0x0" ::: "memory");
    }
    __syncthreads();

    const int wave = tid >> 5;        // 0..7
    const int lane = tid & 31;
    const int half = lane >> 4;       // 0 or 1  (WMMA A-layout lane half)
    const int lr   = lane & 15;       // M index within the 16-row/col group

    const v2f bones = {1.0f, 1.0f};   // all-ones B matrix (4x16, replicated)

    for (int it = 0; it < N_ITERS; ++it) {
        // ================= ROW pass: la[r,:] -= logsumexp over columns =========
        for (int g = 0; g < 2; ++g) {
            const int rowBase = (wave * 2 + g) * 16;
            const int r    = rowBase + lr;
            const int cOff = half * 2;            // lanes 0-15: K=0,1 ; 16-31: K=2,3
            float* rowp = la + r * STRIDE;

            // max over the 128 elements this lane covers, then merge lane halves
            float mx = -3.402823466e38f;
            for (int k0 = 0; k0 < V_DIM; k0 += 4) {
                float a0 = rowp[k0 + cOff];
                float a1 = rowp[k0 + cOff + 1];
                mx = fmaxf(mx, fmaxf(a0, a1));
            }
            mx = fmaxf(mx, __shfl_xor(mx, 16, 32));

            // sum of exp via f32 WMMA: A = exp tile (16x4), B = ones (4x16)
            v8f acc = {};
            for (int k0 = 0; k0 < V_DIM; k0 += 4) {
                v2f a;
                a.x = __expf(rowp[k0 + cOff]     - mx);
                a.y = __expf(rowp[k0 + cOff + 1] - mx);
                acc = __builtin_amdgcn_wmma_f32_16x16x4_f32(
                          false, a, false, bones, (short)0, acc, false, false);
            }
            // D layout: VGPR i = rowsum[rowBase+i] (lanes 0-15) / rowBase+8+i (16-31)
            if (lane == 0) {
                #pragma unroll
                for (int i = 0; i < 8; ++i) red[rowBase + i] = acc[i];
            }
            if (lane == 16) {
                #pragma unroll
                for (int i = 0; i < 8; ++i) red[rowBase + 8 + i] = acc[i];
            }
            __builtin_amdgcn_wave_barrier();      // same-wave LDS ordering fence
            float lse = mx + __logf(red[r]);
            for (int k0 = 0; k0 < V_DIM; k0 += 4) {
                rowp[k0 + cOff]     -= lse;
                rowp[k0 + cOff + 1] -= lse;
            }
        }
        __syncthreads();

        // ================= COL pass: la[:,c] -= logsumexp over rows ============
        for (int g = 0; g < 2; ++g) {
            const int colBase = (wave * 2 + g) * 16;
            const int c    = colBase + lr;
            const int rOff = half * 2;

            float mx = -3.402823466e38f;
            for (int k0 = 0; k0 < V_DIM; k0 += 4) {
                float a0 = la[(k0 + rOff)     * STRIDE + c];
                float a1 = la[(k0 + rOff + 1) * STRIDE + c];
                mx = fmaxf(mx, fmaxf(a0, a1));
            }
            mx = fmaxf(mx, __shfl_xor(mx, 16, 32));

            v8f acc = {};
            for (int k0 = 0; k0 < V_DIM; k0 += 4) {
                v2f a;                             // A = transposed exp tile
                a.x = __expf(la[(k0 + rOff)     * STRIDE + c] - mx);
                a.y = __expf(la[(k0 + rOff + 1) * STRIDE + c] - mx);
                acc = __builtin_amdgcn_wmma_f32_16x16x4_f32(
                          false, a, false, bones, (short)0, acc, false, false);
            }
            if (lane == 0) {
                #pragma unroll
                for (int i = 0; i < 8; ++i) red[colBase + i] = acc[i];
            }
            if (lane == 16) {
                #pragma unroll
                for (int i = 0; i < 8; ++i) red[colBase + 8 + i] = acc[i];
            }
            __builtin_amdgcn_wave_barrier();
            float lse = mx + __logf(red[c]);
            for (int k0 = 0; k0 < V_DIM; k0 += 4) {
                la[(k0 + rOff)     * STRIDE + c] -= lse;
                la[(k0 + rOff + 1) * STRIDE + c] -= lse;
            }
        }
        __syncthreads();
    }

    // out[b,l,w] = exp(la[w, t]) for every (b,l) whose token == t.
    // Each output element is written by exactly one block.
    float val = __expf(la[tid * STRIDE + t]);
    for (int bl = 0; bl < BL; ++bl) {
        if (tokens[bl] == t) {                    // uniform scalar-cached load
            out[bl * V_DIM + tid] = val;          // coalesced 1KB store per match
        }
    }
}

extern "C" void kernel_launch(void* const* d_in, const int* in_sizes, int n_in,
                              void* d_out, int out_size, void* d_ws, size_t ws_size,
                              hipStream_t stream) {
    const int*   tokens = (const int*)d_in[0];     // [8,128] int32
    const float* W      = (const float*)d_in[1];   // [256, 65536] f32
    float*       out    = (float*)d_out;           // [8,128,256] f32
    (void)in_sizes; (void)n_in; (void)d_ws; (void)ws_size; (void)out_size;

    const size_t smemBytes = (size_t)(V_DIM * STRIDE + V_DIM) * sizeof(float); // ~265KB < 320KB
    sinkhorn_per_token_kernel<<<V_DIM, 256, smemBytes, stream>>>(tokens, W, out);
}